// RigidBodySimulator_49185965474108
// MI455X (gfx1250) — compile-verified
//
#include <hip/hip_runtime.h>
#include <math.h>

typedef __attribute__((ext_vector_type(2))) float v2f;
typedef __attribute__((ext_vector_type(8))) float v8f;

#define RB_DT (1.0f/600.0f)
#define BLOCK 256          // 8 waves of 32
#define WPB   8            // waves per block
#define MAXBLK 512

// workspace float-slot layout
#define WS_TICKC 4         // int ticket, contact kernels
#define WS_TICKI 5         // int ticket, inertia kernel
#define WS_T     8         // t[3]
#define WS_Q     11        // q[4]
#define WS_V     15        // v[3]
#define WS_OM    18        // om[3]
#define WS_IREF  21        // I_ref[9]
#define WS_COEF  32        // R10,R11,R12,c0,c1,c2,sdf_off,relv_off
#define WS_PART  64        // 8 floats per block of partial sums

// ---------------- small 3x3 helpers ----------------
__device__ __forceinline__ void inv3(const float* A, float* o) {
  float a=A[0],b=A[1],c=A[2],d=A[3],e=A[4],f=A[5],g=A[6],h=A[7],i=A[8];
  float C0=e*i-f*h, C1=c*h-b*i, C2=b*f-c*e;
  float C3=f*g-d*i, C4=a*i-c*g, C5=c*d-a*f;
  float C6=d*h-e*g, C7=b*g-a*h, C8=a*e-b*d;
  float id=1.0f/(a*C0+b*C3+c*C6);
  o[0]=C0*id;o[1]=C1*id;o[2]=C2*id;o[3]=C3*id;o[4]=C4*id;
  o[5]=C5*id;o[6]=C6*id;o[7]=C7*id;o[8]=C8*id;
}
__device__ __forceinline__ void mm3(const float*A,const float*B,float*C){
  for(int r=0;r<3;r++)for(int c=0;c<3;c++)
    C[r*3+c]=A[r*3]*B[c]+A[r*3+1]*B[3+c]+A[r*3+2]*B[6+c];
}
__device__ __forceinline__ void mmT3(const float*A,const float*B,float*C){ // A @ B^T
  for(int r=0;r<3;r++)for(int c=0;c<3;c++)
    C[r*3+c]=A[r*3]*B[c*3]+A[r*3+1]*B[c*3+1]+A[r*3+2]*B[c*3+2];
}
__device__ __forceinline__ void mv3(const float*A,const float*v,float*o){
  o[0]=A[0]*v[0]+A[1]*v[1]+A[2]*v[2];
  o[1]=A[3]*v[0]+A[4]*v[1]+A[5]*v[2];
  o[2]=A[6]*v[0]+A[7]*v[1]+A[8]*v[2];
}
__device__ __forceinline__ void quat2mat(float w,float x,float y,float z,float* R){
  float inv=1.0f/sqrtf(w*w+x*x+y*y+z*z);
  w*=inv;x*=inv;y*=inv;z*=inv;
  R[0]=1.f-2.f*(y*y+z*z); R[1]=2.f*(x*y-w*z);     R[2]=2.f*(x*z+w*y);
  R[3]=2.f*(x*y+w*z);     R[4]=1.f-2.f*(x*x+z*z); R[5]=2.f*(y*z-w*x);
  R[6]=2.f*(x*z-w*y);     R[7]=2.f*(y*z+w*x);     R[8]=1.f-2.f*(x*x+y*y);
}

// write A-matrix coefficients for the NEXT contact pass
__device__ __forceinline__ void write_coeffs(float* W, const float* R,
                                             float omx, float omz,
                                             float ty, float mcy, float vy){
  W[WS_COEF+0]=R[3]; W[WS_COEF+1]=R[4]; W[WS_COEF+2]=R[5];          // R row1 -> sdf
  W[WS_COEF+3]=omz*R[0]-omx*R[6];                                   // c = om.z*R0 - om.x*R2
  W[WS_COEF+4]=omz*R[1]-omx*R[7];
  W[WS_COEF+5]=omz*R[2]-omx*R[8];
  W[WS_COEF+6]=ty+mcy;                                              // sdf offset
  W[WS_COEF+7]=vy;                                                  // relv offset
}

// ------------- per-step state update (one thread) -------------
__device__ void rb_step_update(float* W, float cnt, float sx, float sy, float sz,
                               const float* kn_p, const float* mu_p,
                               const float* ld_p, const float* ad_p,
                               const float* mc, float invMass, float* outRow)
{
  const float kn=kn_p[0], mu=mu_p[0], ld=ld_p[0], ad=ad_p[0];
  float t0=W[WS_T],t1=W[WS_T+1],t2=W[WS_T+2];
  float q0=W[WS_Q],q1=W[WS_Q+1],q2=W[WS_Q+2],q3=W[WS_Q+3];
  float v0=W[WS_V],v1=W[WS_V+1],v2=W[WS_V+2];
  float o0=W[WS_OM],o1=W[WS_OM+1],o2=W[WS_OM+2];
  float R[9]; quat2mat(q0,q1,q2,q3,R);
  float vo0=v0*ld, vo1=(v1-9.8f*RB_DT)*ld, vo2=v2*ld;
  float oo0=o0*ad, oo1=o1*ad, oo2=o2*ad;
  float denom=fmaxf(cnt,1.f);
  float ri0=sx/denom, ri1=sy/denom, ri2=sz/denom;
  float Ri0=R[0]*ri0+R[1]*ri1+R[2]*ri2;
  float Ri1=R[3]*ri0+R[4]*ri1+R[5]*ri2;
  float Ri2=R[6]*ri0+R[7]*ri1+R[8]*ri2;
  // n = (0,1,0) always (ground plane)
  float vic0=v0+(o1*Ri2-o2*Ri1);
  float vic1=v1+(o2*Ri0-o0*Ri2);
  float vic2=v2+(o0*Ri1-o1*Ri0);
  float nvn=fabsf(vic1);
  float nvt=sqrtf(vic0*vic0+vic2*vic2);
  float alpha=fmaxf(1.f-mu*(1.f+kn)*(nvn/(nvt+1e-6f)),0.f);
  float vin0=alpha*vic0, vin1=-kn*vic1, vin2=alpha*vic2;
  float Iref[9]; for(int i=0;i<9;i++) Iref[i]=W[WS_IREF+i];
  float RI[9]; mm3(R,Iref,RI);
  float Iw[9]; mmT3(RI,R,Iw);
  float Iinv[9]; inv3(Iw,Iinv);
  float Cx[9]={0.f,-Ri2,Ri1, Ri2,0.f,-Ri0, -Ri1,Ri0,0.f};
  float M1[9]; mm3(Cx,Iinv,M1);
  float M2[9]; mm3(M1,Cx,M2);
  float K[9]; for(int i=0;i<9;i++) K[i]=-M2[i];
  K[0]+=invMass; K[4]+=invMass; K[8]+=invMass;
  float rhs[3]={vin0-vic0, vin1-vic1, vin2-vic2};
  float Kinv[9]; inv3(K,Kinv);
  float J[3]; mv3(Kinv,rhs,J);
  float hit=(cnt>0.f)?1.f:0.f;
  vo0+=hit*J[0]*invMass; vo1+=hit*J[1]*invMass; vo2+=hit*J[2]*invMass;
  float CJ[3]; mv3(Cx,J,CJ);
  float IJ[3]; mv3(Iinv,CJ,IJ);
  oo0+=hit*IJ[0]; oo1+=hit*IJ[1]; oo2+=hit*IJ[2];
  float tn0=t0+RB_DT*vo0, tn1=t1+RB_DT*vo1, tn2=t2+RB_DT*vo2;
  float dw=0.5f*RB_DT*(-oo0*q1-oo1*q2-oo2*q3);
  float dx=0.5f*RB_DT*( oo0*q0+oo1*q3-oo2*q2);
  float dy=0.5f*RB_DT*( oo1*q0+oo2*q1-oo0*q3);
  float dz=0.5f*RB_DT*( oo2*q0+oo0*q2-oo1*q1);
  float nq0=q0+dw,nq1=q1+dx,nq2=q2+dy,nq3=q3+dz;
  float qi=1.f/sqrtf(nq0*nq0+nq1*nq1+nq2*nq2+nq3*nq3);
  nq0*=qi;nq1*=qi;nq2*=qi;nq3*=qi;
  W[WS_T]=tn0;W[WS_T+1]=tn1;W[WS_T+2]=tn2;
  W[WS_Q]=nq0;W[WS_Q+1]=nq1;W[WS_Q+2]=nq2;W[WS_Q+3]=nq3;
  W[WS_V]=vo0;W[WS_V+1]=vo1;W[WS_V+2]=vo2;
  W[WS_OM]=oo0;W[WS_OM+1]=oo1;W[WS_OM+2]=oo2;
  outRow[0]=tn0;outRow[1]=tn1;outRow[2]=tn2;
  outRow[3]=nq0;outRow[4]=nq1;outRow[5]=nq2;outRow[6]=nq3;
  float Rn[9]; quat2mat(nq0,nq1,nq2,nq3,Rn);
  write_coeffs(W,Rn,oo0,oo2,tn1,mc[1],vo1);
}

// ---------------- kernels ----------------
__global__ void rb_zero(float* W){
  if (threadIdx.x==0) for (int i=0;i<WS_PART;i++) W[i]=0.f;
}

// one-time inertia reduction + initial state (deterministic last-block pattern)
__global__ void rb_inertia_init(const float* __restrict__ xyz, int N, int nblocks,
                                const float* mc, const float* t0p, const float* q0p,
                                const float* v0p, float* W, int* tick)
{
  __shared__ float red[WPB*8];
  __shared__ int lastFlag;
  const int tid=threadIdx.x, lane=tid&31, wid=tid>>5;
  float s[6]={0,0,0,0,0,0};
  float mcx=mc[0], mcy=mc[1], mcz=mc[2];
  for (int i=blockIdx.x*BLOCK+tid; i<N; i+=nblocks*BLOCK){
    float rx=xyz[3*i]-mcx, ry=xyz[3*i+1]-mcy, rz=xyz[3*i+2]-mcz;
    s[0]+=rx*rx; s[1]+=ry*ry; s[2]+=rz*rz; s[3]+=rx*ry; s[4]+=rx*rz; s[5]+=ry*rz;
  }
  for (int off=16; off>0; off>>=1)
    for (int k=0;k<6;k++) s[k]+=__shfl_xor(s[k],off,32);
  if (lane==0) for (int k=0;k<6;k++) red[wid*8+k]=s[k];
  __syncthreads();
  if (tid==0){
    float p[6]={0,0,0,0,0,0};
    for (int w=0;w<WPB;w++) for (int k=0;k<6;k++) p[k]+=red[w*8+k];
    float* P=W+WS_PART+8*blockIdx.x;
    for (int k=0;k<6;k++) P[k]=p[k];
    __threadfence();
    lastFlag=(atomicAdd(tick,1)==nblocks-1);
  }
  __syncthreads();
  if (lastFlag){
    __threadfence();
    float p[6]={0,0,0,0,0,0};
    for (int b=tid; b<nblocks; b+=BLOCK){
      const float* P=W+WS_PART+8*b;
      for (int k=0;k<6;k++) p[k]+=P[k];
    }
    for (int off=16; off>0; off>>=1)
      for (int k=0;k<6;k++) p[k]+=__shfl_xor(p[k],off,32);
    if (lane==0) for (int k=0;k<6;k++) red[wid*8+k]=p[k];
    __syncthreads();
    if (tid==0){
      float S[6]={0,0,0,0,0,0};
      for (int w=0;w<WPB;w++) for (int k=0;k<6;k++) S[k]+=red[w*8+k];
      float tr=S[0]+S[1]+S[2];
      W[WS_IREF+0]=tr-S[0]; W[WS_IREF+1]=-S[3];    W[WS_IREF+2]=-S[4];
      W[WS_IREF+3]=-S[3];   W[WS_IREF+4]=tr-S[1];  W[WS_IREF+5]=-S[5];
      W[WS_IREF+6]=-S[4];   W[WS_IREF+7]=-S[5];    W[WS_IREF+8]=tr-S[2];
      // initial state
      W[WS_T]=t0p[0];W[WS_T+1]=t0p[1];W[WS_T+2]=t0p[2];
      float a=q0p[0],b=q0p[1],c=q0p[2],d=q0p[3];
      float qi=1.f/sqrtf(a*a+b*b+c*c+d*d);
      W[WS_Q]=a*qi;W[WS_Q+1]=b*qi;W[WS_Q+2]=c*qi;W[WS_Q+3]=d*qi;
      W[WS_V]=v0p[0];W[WS_V+1]=v0p[1];W[WS_V+2]=v0p[2];
      W[WS_OM]=0.f;W[WS_OM+1]=0.f;W[WS_OM+2]=0.f;
      float R[9]; quat2mat(W[WS_Q],W[WS_Q+1],W[WS_Q+2],W[WS_Q+3],R);
      write_coeffs(W,R,0.f,0.f,t0p[1],mc[1],v0p[1]);   // om = 0 initially
      atomicExch(tick,0);
    }
  }
}

// per-step: WMMA contact detection + masked reduction; last block updates state
__global__ void rb_contact_step(const float* __restrict__ xyz, int N, int ntiles,
                                int nblocks, const float* kn_p, const float* mu_p,
                                const float* ld_p, const float* ad_p, const float* mc,
                                float invMass, float* W, int* tick, float* outRow)
{
  __shared__ float red[WPB*8];
  __shared__ int lastFlag;
  const int tid=threadIdx.x, lane=tid&31, wid=tid>>5;
  const int col=lane&15, half=lane>>4;

  // A-matrix: row0 = R[1,:] (sdf weights), row1 = om.z*R[0,:]-om.x*R[2,:] (rel-vel weights)
  const float R10=W[WS_COEF],R11=W[WS_COEF+1],R12=W[WS_COEF+2];
  const float c0=W[WS_COEF+3],c1=W[WS_COEF+4],c2=W[WS_COEF+5];
  const float sdf_off=W[WS_COEF+6], relv_off=W[WS_COEF+7];

  v2f a; a.x=0.f; a.y=0.f;            // A 16x4 f32: lanes 0-15 = K0,K1; lanes 16-31 = K2,K3
  if (col==0){ a.x = half? R12 : R10; a.y = half? 0.f : R11; }
  else if (col==1){ a.x = half? c2 : c0; a.y = half? 0.f : c1; }

  float cnt=0.f, sx=0.f, sy=0.f, sz=0.f;

  for (int tile=blockIdx.x*WPB+wid; tile<ntiles; tile+=nblocks*WPB){
    int nextTile = tile + nblocks*WPB;
    if (nextTile < ntiles)
      __builtin_prefetch(xyz + 3*(size_t)(nextTile*16 + col), 0, 3);
    int vidx=tile*16+col;
    bool valid=vidx<N;
    const float* vp=xyz+3*(valid?vidx:0);
    float b0,b1;
    if (half==0){ b0=vp[0]; b1=vp[1]; } else { b0=vp[2]; b1=0.f; }
    if (!valid){ b0=0.f; b1=0.f; }
    v2f bv; bv.x=b0; bv.y=b1;          // B 4x16 f32: columns = vertices
    v8f cz={0.f,0.f,0.f,0.f,0.f,0.f,0.f,0.f};
    v8f d=__builtin_amdgcn_wmma_f32_16x16x4_f32(false,a,false,bv,(short)0,cz,false,false);
    float sdfp=d[0]+sdf_off;           // lane n holds D[0][n], D[1][n]
    float relv=d[1]+relv_off;
    float m=(valid && sdfp<0.f && relv<0.f)?1.f:0.f;
    float mo=__shfl_xor(m,16,32);      // hand mask to the z-holding half-wave
    if (half==0){ cnt+=m; sx+=m*b0; sy+=m*b1; }
    else        { sz+=mo*b0; }
  }
  for (int off=16; off>0; off>>=1){
    cnt+=__shfl_xor(cnt,off,32); sx+=__shfl_xor(sx,off,32);
    sy +=__shfl_xor(sy ,off,32); sz+=__shfl_xor(sz,off,32);
  }
  if (lane==0){ red[wid*8]=cnt; red[wid*8+1]=sx; red[wid*8+2]=sy; red[wid*8+3]=sz; }
  __syncthreads();
  if (tid==0){
    float p[4]={0,0,0,0};
    for (int w=0;w<WPB;w++){ p[0]+=red[w*8]; p[1]+=red[w*8+1]; p[2]+=red[w*8+2]; p[3]+=red[w*8+3]; }
    float* P=W+WS_PART+8*blockIdx.x;
    P[0]=p[0];P[1]=p[1];P[2]=p[2];P[3]=p[3];
    __threadfence();
    lastFlag=(atomicAdd(tick,1)==nblocks-1);
  }
  __syncthreads();
  if (lastFlag){
    __threadfence();
    float p[4]={0,0,0,0};
    for (int b=tid; b<nblocks; b+=BLOCK){
      const float* P=W+WS_PART+8*b;
      p[0]+=P[0]; p[1]+=P[1]; p[2]+=P[2]; p[3]+=P[3];
    }
    for (int off=16; off>0; off>>=1)
      for (int k=0;k<4;k++) p[k]+=__shfl_xor(p[k],off,32);
    if (lane==0){ red[wid*8]=p[0]; red[wid*8+1]=p[1]; red[wid*8+2]=p[2]; red[wid*8+3]=p[3]; }
    __syncthreads();
    if (tid==0){
      float C=0,X=0,Y=0,Z=0;
      for (int w=0;w<WPB;w++){ C+=red[w*8]; X+=red[w*8+1]; Y+=red[w*8+2]; Z+=red[w*8+3]; }
      rb_step_update(W,C,X,Y,Z,kn_p,mu_p,ld_p,ad_p,mc,invMass,outRow);
      atomicExch(tick,0);
    }
  }
}

static inline int imin_(int a,int b){return a<b?a:b;}

extern "C" void kernel_launch(void* const* d_in, const int* in_sizes, int n_in,
                              void* d_out, int out_size, void* d_ws, size_t ws_size,
                              hipStream_t stream) {
  const float* x  = (const float*)d_in[0];
  const float* mc = (const float*)d_in[1];
  const float* t0 = (const float*)d_in[2];
  const float* q0 = (const float*)d_in[3];
  const float* v0 = (const float*)d_in[4];
  const float* kn = (const float*)d_in[5];
  const float* mu = (const float*)d_in[6];
  const float* ld = (const float*)d_in[7];
  const float* ad = (const float*)d_in[8];
  int N = in_sizes[0]/3;
  int steps = out_size/7;
  float* out = (float*)d_out;
  float* W = (float*)d_ws;
  int* tickC = (int*)d_ws + WS_TICKC;
  int* tickI = (int*)d_ws + WS_TICKI;
  float invMass = 1.0f/(float)N;

  int ntiles = (N+15)/16;
  int wsBlocks = (int)((ws_size/sizeof(float) - WS_PART)/8);
  if (wsBlocks < 1) wsBlocks = 1;
  int nbC = imin_(imin_(MAXBLK,(ntiles+WPB-1)/WPB), wsBlocks);
  if (nbC < 1) nbC = 1;
  int nbI = imin_(imin_(MAXBLK,(N+BLOCK-1)/BLOCK), wsBlocks);
  if (nbI < 1) nbI = 1;

  rb_zero<<<1,32,0,stream>>>(W);
  rb_inertia_init<<<nbI,BLOCK,0,stream>>>(x,N,nbI,mc,t0,q0,v0,W,tickI);
  for (int s=0;s<steps;s++)
    rb_contact_step<<<nbC,BLOCK,0,stream>>>(x,N,ntiles,nbC,kn,mu,ld,ad,mc,
                                            invMass,W,tickC,out+7*s);
}